// GeneSignatureDecoder_71201967833468
// MI455X (gfx1250) — compile-verified
//
#include <hip/hip_runtime.h>

typedef __attribute__((ext_vector_type(2))) float v2f;
typedef __attribute__((ext_vector_type(4))) float v4f;
typedef __attribute__((ext_vector_type(8))) float v8f;

#define N_BATCH 256
#define SIG     10
#define NC      23      // chromosomes
#define NCP     24      // padded K (multiple of 4)
#define NG      20000   // genes
#define NF      32      // hidden
#define BN_EPS  1e-5f
#define G_TILE  128     // 8 waves * 16 genes
#define N_TILE  16
#define NBLK_G  ((NG + G_TILE - 1) / G_TILE)   // 157
#define NG_PAD  (NBLK_G * G_TILE)              // 20096, zero-padded rows

// ---------------------------------------------------------------------------
// Kernel 1: h = relu(x @ lin_w^T + lin_b); batch-norm with batch stats;
// write hn padded (256 x 24), column 23 zeroed, to workspace.
// Single block, 256 threads (thread t == batch row t).
// ---------------------------------------------------------------------------
__global__ __launch_bounds__(256) void gsd_prep_kernel(
    const float* __restrict__ x, const float* __restrict__ lin_w,
    const float* __restrict__ lin_b, const float* __restrict__ bn_gamma,
    const float* __restrict__ bn_beta, float* __restrict__ hn)
{
    __shared__ float slw[NC * SIG];
    __shared__ float slb[NC], sgam[NC], sbet[NC], ssum[NC], ssq[NC];
    const int t = threadIdx.x;
    if (t < NC * SIG) slw[t] = lin_w[t];
    if (t < NC) {
        slb[t]  = lin_b[t];
        sgam[t] = bn_gamma[t];
        sbet[t] = bn_beta[t];
        ssum[t] = 0.0f;
        ssq[t]  = 0.0f;
    }
    __syncthreads();

    float xr[SIG];
#pragma unroll
    for (int i = 0; i < SIG; ++i) xr[i] = x[t * SIG + i];

    float h[NC];
#pragma unroll
    for (int c = 0; c < NC; ++c) {
        float acc = slb[c];
#pragma unroll
        for (int i = 0; i < SIG; ++i) acc = fmaf(xr[i], slw[c * SIG + i], acc);
        h[c] = fmaxf(acc, 0.0f);
        atomicAdd(&ssum[c], h[c]);            // ds_add_f32
        atomicAdd(&ssq[c],  h[c] * h[c]);
    }
    __syncthreads();

#pragma unroll
    for (int c = 0; c < NC; ++c) {
        const float mu  = ssum[c] * (1.0f / N_BATCH);
        const float var = ssq[c] * (1.0f / N_BATCH) - mu * mu;  // biased, torch BN
        hn[t * NCP + c] = (h[c] - mu) * rsqrtf(var + BN_EPS) * sgam[c] + sbet[c];
    }
    hn[t * NCP + NC] = 0.0f;  // K padding
}

// ---------------------------------------------------------------------------
// Kernel 2: repack gene_w (20000 x 23) into gpad (20096 x 24), zero-padded in
// both K (c=23) and rows (>=20000), and fuse r[g] = sum_c gene_w[g,c].
// Removes all bounds checks from the decode hot loop: B fragments become
// unconditional 8B-aligned b64 loads.
// ---------------------------------------------------------------------------
__global__ __launch_bounds__(256) void gsd_pad_kernel(
    const float* __restrict__ gmat, float* __restrict__ gpad,
    float* __restrict__ rsum)
{
    const int g = blockIdx.x * blockDim.x + threadIdx.x;  // one thread per row
    if (g >= NG_PAD) return;
    float s = 0.0f;
    if (g < NG) {
#pragma unroll
        for (int c = 0; c < NC; ++c) {
            const float v = gmat[g * NC + c];
            gpad[g * NCP + c] = v;
            s += v;
        }
        gpad[g * NCP + NC] = 0.0f;
    } else {
#pragma unroll
        for (int c = 0; c < NCP; ++c) gpad[g * NCP + c] = 0.0f;
    }
    rsum[g] = s;
}

// ---------------------------------------------------------------------------
// Kernel 3: s = hn @ gene_w^T via V_WMMA_F32_16X16X4_F32 (M=batch, N=gene,
// K=24 in 6 steps), then out[n,g,f] = s[n,g]*cw[f] + r[g]*cb[f]  (the Conv1d
// k=1 axis is rank-1). Block = 256 threads = 8 wave32; wave w owns genes
// [g_base + 16w, +16). Grid = (157, 16).
// HBM-write bound: 655 MB fp32 -> non-temporal stores so the 192 MB L2 stays
// reserved for the reused gpad/hn/rsum operands.
// ---------------------------------------------------------------------------
__global__ __launch_bounds__(256) void gsd_decode_kernel(
    const float* __restrict__ gpad, const float* __restrict__ hn,
    const float* __restrict__ rsum, const float* __restrict__ conv_w,
    const float* __restrict__ conv_b, float* __restrict__ out)
{
    __shared__ float cw[NF], cb[NF];
    const int t = threadIdx.x;
    if (t < NF) { cw[t] = conv_w[t]; cb[t] = conv_b[t]; }
    __syncthreads();

    const int wave = t >> 5;
    const int lane = t & 31;
    const int half = lane >> 4;   // 0: lanes 0-15 (K 0,1) | 1: lanes 16-31 (K 2,3)
    const int l16  = lane & 15;

    const int g_base  = blockIdx.x * G_TILE + wave * 16;
    const int n_base  = blockIdx.y * N_TILE;
    const int my_gene = g_base + l16;            // B-matrix column / C-matrix N

    // A fragment: hn row (M = n_base + l16), stride 24; B fragment: padded
    // gene row (always in-bounds). Both 8B-aligned -> global_load_b64.
    const float* arow = hn   + (n_base + l16) * NCP + half * 2;
    const float* brow = gpad + my_gene * NCP + half * 2;
    __builtin_prefetch(brow, 0, 1);              // global_prefetch_b8

    v8f acc = {};
#pragma unroll
    for (int ks = 0; ks < 6; ++ks) {
        const v2f a = *reinterpret_cast<const v2f*>(arow + ks * 4);
        const v2f b = *reinterpret_cast<const v2f*>(brow + ks * 4);
        // D = A(16x4) * B(4x16) + C ; fp32 throughout
        acc = __builtin_amdgcn_wmma_f32_16x16x4_f32(
            false, a, false, b, (short)0, acc, false, false);
    }

    if (my_gene >= NG) return;  // tail genes (padding rows): nothing to write

    const float r = rsum[my_gene];

    // C/D layout: VGPR j holds M = j + 8*half, N = l16.
#pragma unroll
    for (int j = 0; j < 8; ++j) {
        const int   m = j + half * 8;
        const float s = acc[j];
        float* optr = out + ((size_t)(n_base + m) * NG + my_gene) * NF;
#pragma unroll
        for (int f4 = 0; f4 < 8; ++f4) {
            v4f v;
            v.x = fmaf(s, cw[f4 * 4 + 0], r * cb[f4 * 4 + 0]);
            v.y = fmaf(s, cw[f4 * 4 + 1], r * cb[f4 * 4 + 1]);
            v.z = fmaf(s, cw[f4 * 4 + 2], r * cb[f4 * 4 + 2]);
            v.w = fmaf(s, cw[f4 * 4 + 3], r * cb[f4 * 4 + 3]);
            // Streaming output: global_store_b128 th:TH_STORE_NT
            __builtin_nontemporal_store(v, reinterpret_cast<v4f*>(optr + f4 * 4));
        }
    }
}

// ---------------------------------------------------------------------------
extern "C" void kernel_launch(void* const* d_in, const int* in_sizes, int n_in,
                              void* d_out, int out_size, void* d_ws, size_t ws_size,
                              hipStream_t stream) {
    (void)in_sizes; (void)n_in; (void)out_size; (void)ws_size;

    const float* x        = (const float*)d_in[0];
    const float* lin_w    = (const float*)d_in[1];
    const float* lin_b    = (const float*)d_in[2];
    const float* bn_gamma = (const float*)d_in[3];
    const float* bn_beta  = (const float*)d_in[4];
    const float* conv_w   = (const float*)d_in[5];
    const float* conv_b   = (const float*)d_in[6];
    const float* gene_w   = (const float*)d_in[7];
    // d_in[8] == mask is redundant: gene_w * (gene_w != 0) == gene_w.

    float* out  = (float*)d_out;
    float* hn   = (float*)d_ws;                  // 256*24 floats
    float* rsum = hn + N_BATCH * NCP;            // 20096 floats
    float* gpad = rsum + NG_PAD;                 // 20096*24 floats (~1.9 MB)

    gsd_prep_kernel<<<1, 256, 0, stream>>>(x, lin_w, lin_b, bn_gamma, bn_beta, hn);
    gsd_pad_kernel<<<(NG_PAD + 255) / 256, 256, 0, stream>>>(gene_w, gpad, rsum);

    dim3 grid(NBLK_G, N_BATCH / N_TILE);
    gsd_decode_kernel<<<grid, 256, 0, stream>>>(gpad, hn, rsum, conv_w, conv_b, out);
}